// HealpixUNet_63702954934345
// MI455X (gfx1250) — compile-verified
//
#include <hip/hip_runtime.h>

typedef float v2f __attribute__((ext_vector_type(2)));
typedef float v4f __attribute__((ext_vector_type(4)));
typedef float v8f __attribute__((ext_vector_type(8)));

// ---------------------------------------------------------------------------
// Fast WMMA f32 GEMM: Y[M x Nc] (+)= X[M x K] @ W[K x Nc]
// Requires: M % 64 == 0, K % 4 == 0, Nc % 16 == 0.
// Each wave computes FOUR 16x16 output tiles stacked along M (64 rows x 16 cols),
// reusing the B fragment across the 4 wmmas per K-step. No guards -> no exec
// manipulation in the hot loop.
// A (16x4): lane<16 holds K=kk,kk+1 ; lane>=16 holds K=kk+2,kk+3 ; M=lane&15
// B (4x16): same K split across lane halves; N=lane&15
// C/D: vgpr r -> M = r + (lane>=16 ? 8:0), N = lane&15
// ---------------------------------------------------------------------------
__global__ void wmma_mm_fast(const float* __restrict__ X, const float* __restrict__ W,
                             float* __restrict__ Y, int M, int K, int Nc, int accum)
{
    const int wave = blockIdx.x * (blockDim.x >> 5) + (threadIdx.x >> 5);
    const int tilesN = Nc >> 4;
    const int mblocks = M >> 6;                 // 64 rows per wave
    if (wave >= mblocks * tilesN) return;       // wave-uniform exit
    const int m0 = (wave / tilesN) << 6;
    const int n0 = (wave % tilesN) << 4;
    const int lane = threadIdx.x & 31;
    const int half = lane >> 4;                 // 0 or 1
    const int l15 = lane & 15;

    const int cn = n0 + l15;
    const int cmBase = m0 + (half ? 8 : 0);

    v8f acc[4];
    if (accum) {
#pragma unroll
        for (int t = 0; t < 4; ++t) {
            const int cm = cmBase + t * 16;
#pragma unroll
            for (int r = 0; r < 8; ++r)
                acc[t][r] = Y[(cm + r) * Nc + cn];
        }
    } else {
#pragma unroll
        for (int t = 0; t < 4; ++t)
#pragma unroll
            for (int r = 0; r < 8; ++r)
                acc[t][r] = 0.0f;
    }

    const int am = m0 + l15;                    // A row for tile t: am + t*16
    const int koff = half * 2;
    for (int kk = 0; kk < K; kk += 4) {
        const float* wp = W + (kk + koff) * Nc + cn;
        v2f b;
        b.x = wp[0];
        b.y = wp[Nc];
#pragma unroll
        for (int t = 0; t < 4; ++t) {
            const float* xp = X + (am + t * 16) * K + kk + koff;
            v2f a = *(const v2f*)xp;            // contiguous 8B per lane
            acc[t] = __builtin_amdgcn_wmma_f32_16x16x4_f32(
                false, a, false, b, (short)0, acc[t], false, false);
        }
    }

#pragma unroll
    for (int t = 0; t < 4; ++t) {
        const int cm = cmBase + t * 16;
#pragma unroll
        for (int r = 0; r < 8; ++r)
            Y[(cm + r) * Nc + cn] = acc[t][r];
    }
}

// ---------------------------------------------------------------------------
// Edge-case WMMA GEMM (K not multiple of 4 or Nc not multiple of 16).
// Single 16x16 tile per wave. Guards implemented with clamped indices +
// multiplicative masks (v_cndmask) so loads stay unconditional (no exec churn).
// ---------------------------------------------------------------------------
__global__ void wmma_mm_edge(const float* __restrict__ X, const float* __restrict__ W,
                             float* __restrict__ Y, int M, int K, int Nc, int accum)
{
    const int tile = blockIdx.x * (blockDim.x >> 5) + (threadIdx.x >> 5);
    const int tilesN = (Nc + 15) >> 4;
    if (tile >= (M >> 4) * tilesN) return;      // wave-uniform exit
    const int m0 = (tile / tilesN) << 4;
    const int n0 = (tile % tilesN) << 4;
    const int lane = threadIdx.x & 31;
    const int half = lane >> 4;
    const int l15 = lane & 15;

    const int cm = m0 + (half ? 8 : 0);
    const int cn = n0 + l15;
    const int cnc = cn < Nc ? cn : Nc - 1;      // clamped column
    const float nmask = cn < Nc ? 1.0f : 0.0f;

    v8f c;
#pragma unroll
    for (int r = 0; r < 8; ++r)
        c[r] = accum ? Y[(cm + r) * Nc + cnc] * nmask : 0.0f;

    const int am = m0 + l15;
    for (int kk = 0; kk < K; kk += 4) {
        const int ka = kk + half * 2;
        const int ka0 = ka < K ? ka : K - 1;
        const int ka1 = ka + 1 < K ? ka + 1 : K - 1;
        const float k0m = ka < K ? 1.0f : 0.0f;
        const float k1m = ka + 1 < K ? 1.0f : 0.0f;
        v2f a, b;
        a.x = X[am * K + ka0] * k0m;
        a.y = X[am * K + ka1] * k1m;
        b.x = W[ka0 * Nc + cnc] * (k0m * nmask);
        b.y = W[ka1 * Nc + cnc] * (k1m * nmask);
        c = __builtin_amdgcn_wmma_f32_16x16x4_f32(false, a, false, b, (short)0, c,
                                                  false, false);
    }
#pragma unroll
    for (int r = 0; r < 8; ++r)
        if (cn < Nc) Y[(cm + r) * Nc + cn] = c[r];
}

// ---------------------------------------------------------------------------
// SpMV for the fixed-degree (9 nnz/row, row-sorted) Laplacian, 4 channels/thread:
//   xout[b,n,c] = alpha * sum_j vals[9n+j] * xin[b, cols[9n+j], c] + beta*xsub[b,n,c]
// ---------------------------------------------------------------------------
__global__ void spmv4_k(const int* __restrict__ cols, const float* __restrict__ vals,
                        const float* __restrict__ xin, const float* __restrict__ xsub,
                        float* __restrict__ xout, int Np, int C4, float alpha, float beta)
{
    int i = blockIdx.x * blockDim.x + threadIdx.x;
    int tot = 2 * Np * C4;
    if (i >= tot) return;
    int c = i % C4;
    int rem = i / C4;
    int n = rem % Np;
    int b = rem / Np;
    const v4f* xin4 = (const v4f*)xin;
    int bbase = b * Np * C4;
    int e0 = n * 9;
    v4f acc = {0.0f, 0.0f, 0.0f, 0.0f};
#pragma unroll
    for (int j = 0; j < 9; ++j) {
        float v = vals[e0 + j];
        v4f t = xin4[bbase + cols[e0 + j] * C4 + c];
        acc += v * t;
    }
    v4f r = alpha * acc;
    if (beta != 0.0f) r += beta * ((const v4f*)xsub)[i];
    ((v4f*)xout)[i] = r;
}

// Scalar SpMV fallback (used only for the 3-channel network input).
__global__ void spmv_k(const int* __restrict__ cols, const float* __restrict__ vals,
                       const float* __restrict__ xin, const float* __restrict__ xsub,
                       float* __restrict__ xout, int Np, int C, float alpha, float beta)
{
    int i = blockIdx.x * blockDim.x + threadIdx.x;
    int tot = 2 * Np * C;
    if (i >= tot) return;
    int c = i % C;
    int rem = i / C;
    int n = rem % Np;
    int b = rem / Np;
    int bbase = b * Np * C;
    int e0 = n * 9;
    float acc = 0.0f;
#pragma unroll
    for (int j = 0; j < 9; ++j)
        acc += vals[e0 + j] * xin[bbase + cols[e0 + j] * C + c];
    float r = alpha * acc;
    if (beta != 0.0f) r += beta * xsub[i];
    xout[i] = r;
}

__global__ void zero_k(float* __restrict__ p, int n)
{
    int i = blockIdx.x * blockDim.x + threadIdx.x;
    if (i < n) p[i] = 0.0f;
}

// Per-channel sum and sum-of-squares over rows = B*N (batchnorm batch stats).
__global__ void stats_k(const float* __restrict__ y, float* __restrict__ sum,
                        float* __restrict__ sumsq, int rows, int C, int chunk)
{
    int c = threadIdx.x;
    if (c >= C) return;
    int r0 = blockIdx.x * chunk;
    int r1 = r0 + chunk; if (r1 > rows) r1 = rows;
    float s = 0.0f, ss = 0.0f;
    for (int r = r0; r < r1; ++r) {
        float v = y[r * C + c];
        s += v; ss += v * v;
    }
    atomicAdd(&sum[c], s);
    atomicAdd(&sumsq[c], ss);
}

// In-place batchnorm (batch stats) + ReLU.
__global__ void bnrelu_k(float* __restrict__ y, const float* __restrict__ sum,
                         const float* __restrict__ sumsq, const float* __restrict__ g,
                         const float* __restrict__ be, int rows, int C)
{
    int i = blockIdx.x * blockDim.x + threadIdx.x;
    if (i >= rows * C) return;
    int c = i % C;
    float inv = 1.0f / (float)rows;
    float m = sum[c] * inv;
    float v = sumsq[c] * inv - m * m;
    float r = (y[i] - m) * rsqrtf(v + 1e-5f) * g[c] + be[c];
    y[i] = r > 0.0f ? r : 0.0f;
}

// 4:1 maxpool over NEST-ordered children (contiguous groups of 4), with argmax.
__global__ void maxpool_k(const float* __restrict__ in, float* __restrict__ out,
                          int* __restrict__ idx, int Nin, int C)
{
    int Nout = Nin >> 2;
    int i = blockIdx.x * blockDim.x + threadIdx.x;
    int tot = 2 * Nout * C;
    if (i >= tot) return;
    int c = i % C;
    int rem = i / C;
    int m = rem % Nout;
    int b = rem / Nout;
    int base = (b * Nin + 4 * m) * C + c;
    float best = in[base]; int bj = 0;
#pragma unroll
    for (int j = 1; j < 4; ++j) {
        float v = in[base + j * C];
        if (v > best) { best = v; bj = j; }
    }
    out[i] = best;
    idx[i] = bj;
}

// Max-unpool into channels [0, Cin) of a wider concat buffer (stride Ccat).
__global__ void unpool_k(const float* __restrict__ in, const int* __restrict__ idx,
                         float* __restrict__ out, int Nout, int Cin, int Ccat)
{
    int i = blockIdx.x * blockDim.x + threadIdx.x;
    int tot = 2 * Nout * Cin;
    if (i >= tot) return;
    int c = i % Cin;
    int rem = i / Cin;
    int n = rem % Nout;
    int b = rem / Nout;
    int m = n >> 2, j = n & 3;
    int src = (b * (Nout >> 2) + m) * Cin + c;
    float v = (idx[src] == j) ? in[src] : 0.0f;
    out[(b * Nout + n) * Ccat + c] = v;
}

// Copy skip tensor into channels [coff, coff+Cs) of the concat buffer.
__global__ void copychan_k(const float* __restrict__ src, float* __restrict__ dst,
                           int N, int Cs, int Ccat, int coff)
{
    int i = blockIdx.x * blockDim.x + threadIdx.x;
    int tot = 2 * N * Cs;
    if (i >= tot) return;
    int c = i % Cs;
    int rem = i / Cs;
    int n = rem % N;
    int b = rem / N;
    dst[(b * N + n) * Ccat + coff + c] = src[i];
}

__global__ void addbias_k(float* __restrict__ y, const float* __restrict__ b, int tot)
{
    int i = blockIdx.x * blockDim.x + threadIdx.x;
    if (i < tot) y[i] += b[0];
}

// ---------------------------------------------------------------------------
// Host orchestration
// ---------------------------------------------------------------------------
extern "C" void kernel_launch(void* const* d_in, const int* in_sizes, int n_in,
                              void* d_out, int out_size, void* d_ws, size_t ws_size,
                              hipStream_t stream)
{
    (void)in_sizes; (void)n_in; (void)out_size; (void)ws_size;
    static const int NP[4] = {49152, 12288, 3072, 768};

    const float* x = (const float*)d_in[0];
    const int* cols[4]; const float* vals[4];
    for (int l = 0; l < 4; ++l) {
        cols[l] = (const int*)d_in[1 + 3 * l + 1];
        vals[l] = (const float*)d_in[1 + 3 * l + 2];
    }
    struct Prm { const float *W, *b, *g, *be; };
    Prm prm[13];
    int p = 13;
    for (int i = 0; i < 12; ++i) {
        prm[i].W  = (const float*)d_in[p++];
        prm[i].b  = (const float*)d_in[p++];
        prm[i].g  = (const float*)d_in[p++];
        prm[i].be = (const float*)d_in[p++];
    }
    prm[12].W = (const float*)d_in[p++];
    prm[12].b = (const float*)d_in[p++];
    // prm: 0 enc0a, 1 enc0b, 2 down0, 3 down1, 4 down2, 5 bott,
    //      6 pre0, 7 post0, 8 pre1, 9 post1, 10 pre2, 11 post2, 12 outc

    // --- workspace bump allocator ---
    char* wsp = (char*)d_ws;
    size_t off = 0;
    auto alloc = [&](size_t bytes) -> void* {
        void* r = wsp + off;
        off += (bytes + 255) & ~(size_t)255;
        return r;
    };
    const size_t BIG = (size_t)2 * 49152 * 96;   // largest (B,N,C) tensor in floats
    float* bufA = (float*)alloc(BIG * 4);
    float* bufB = (float*)alloc(BIG * 4);
    float* S0   = (float*)alloc(BIG * 4);        // Cheb recurrence scratch
    float* S1   = (float*)alloc(BIG * 4);
    float* S2   = (float*)alloc(BIG * 4);
    float* SK0  = (float*)alloc((size_t)2 * 49152 * 32 * 4);
    float* SK1  = (float*)alloc((size_t)2 * 12288 * 64 * 4);
    float* SK2  = (float*)alloc((size_t)2 * 3072 * 128 * 4);
    int* idx0   = (int*)alloc((size_t)2 * 12288 * 64 * 4);
    int* idx1   = (int*)alloc((size_t)2 * 3072 * 128 * 4);
    int* idx2   = (int*)alloc((size_t)2 * 768 * 256 * 4);
    float* sums = (float*)alloc(512 * 4);        // [0..255]=sum, [256..511]=sumsq

    auto mm = [&](float* Y, const float* X, const float* Wp, int M, int K, int Nc, int acc) {
        if ((K & 3) == 0 && (Nc & 15) == 0) {
            int waves = (M >> 6) * (Nc >> 4);
            int wpb = 4;                          // 128 threads = 4 waves
            int nb = (waves + wpb - 1) / wpb;
            wmma_mm_fast<<<dim3(nb), dim3(wpb * 32), 0, stream>>>(X, Wp, Y, M, K, Nc, acc);
        } else {
            int tiles = (M >> 4) * ((Nc + 15) >> 4);
            int wpb = 8;
            int nb = (tiles + wpb - 1) / wpb;
            wmma_mm_edge<<<dim3(nb), dim3(wpb * 32), 0, stream>>>(X, Wp, Y, M, K, Nc, acc);
        }
    };
    auto spmv = [&](int lvl, const float* xin, const float* xsub, float* xout,
                    int C, float alpha, float beta) {
        if ((C & 3) == 0) {
            int C4 = C >> 2;
            int tot = 2 * NP[lvl] * C4;
            spmv4_k<<<dim3((tot + 255) / 256), dim3(256), 0, stream>>>(
                cols[lvl], vals[lvl], xin, xsub, xout, NP[lvl], C4, alpha, beta);
        } else {
            int tot = 2 * NP[lvl] * C;
            spmv_k<<<dim3((tot + 255) / 256), dim3(256), 0, stream>>>(
                cols[lvl], vals[lvl], xin, xsub, xout, NP[lvl], C, alpha, beta);
        }
    };
    // y = sum_{k=0..4} T_k(L) in @ W_k   (bias handled by caller)
    auto cheb = [&](int lvl, const float* in, int Cin, const float* W, int Cout, float* y) {
        int M = 2 * NP[lvl];
        size_t ws = (size_t)Cin * Cout;
        mm(y, in, W, M, Cin, Cout, 0);               // T0 = in
        spmv(lvl, in, in, S0, Cin, 1.0f, 0.0f);      // T1 = L in
        mm(y, S0, W + 1 * ws, M, Cin, Cout, 1);
        spmv(lvl, S0, in, S1, Cin, 2.0f, -1.0f);     // T2 = 2 L T1 - T0
        mm(y, S1, W + 2 * ws, M, Cin, Cout, 1);
        spmv(lvl, S1, S0, S2, Cin, 2.0f, -1.0f);     // T3
        mm(y, S2, W + 3 * ws, M, Cin, Cout, 1);
        spmv(lvl, S2, S1, S0, Cin, 2.0f, -1.0f);     // T4 (reuse S0)
        mm(y, S0, W + 4 * ws, M, Cin, Cout, 1);
    };
    // Cheb conv -> batchnorm (batch stats; zero bias cancels in BN) -> ReLU
    auto block = [&](int lvl, const float* in, int Cin, const Prm& pp, int Cout, float* y) {
        cheb(lvl, in, Cin, pp.W, Cout, y);
        zero_k<<<dim3(2), dim3(256), 0, stream>>>(sums, 512);
        int rows = 2 * NP[lvl];
        int chunk = 512;
        int nb = (rows + chunk - 1) / chunk;
        int bt = ((Cout + 63) / 64) * 64;
        stats_k<<<dim3(nb), dim3(bt), 0, stream>>>(y, sums, sums + 256, rows, Cout, chunk);
        int tot = rows * Cout;
        bnrelu_k<<<dim3((tot + 255) / 256), dim3(256), 0, stream>>>(
            y, sums, sums + 256, pp.g, pp.be, rows, Cout);
    };
    auto pool = [&](const float* in, float* out, int* idx, int lvl_in, int C) {
        int tot = 2 * (NP[lvl_in] >> 2) * C;
        maxpool_k<<<dim3((tot + 255) / 256), dim3(256), 0, stream>>>(in, out, idx,
                                                                     NP[lvl_in], C);
    };
    auto unpool_cat = [&](const float* in, const int* idx, const float* skip,
                          float* out, int lvl_out, int Cpre, int Cskip) {
        int Ccat = Cpre + Cskip;
        int tot1 = 2 * NP[lvl_out] * Cpre;
        unpool_k<<<dim3((tot1 + 255) / 256), dim3(256), 0, stream>>>(
            in, idx, out, NP[lvl_out], Cpre, Ccat);
        int tot2 = 2 * NP[lvl_out] * Cskip;
        copychan_k<<<dim3((tot2 + 255) / 256), dim3(256), 0, stream>>>(
            skip, out, NP[lvl_out], Cskip, Ccat, Cpre);
    };

    // ---------------- forward pass ----------------
    block(0, x,    3,  prm[0], 16,  bufA);          // enc0a
    block(0, bufA, 16, prm[1], 32,  SK0);           // enc0b -> skip x0
    block(0, SK0,  32, prm[2], 64,  bufB);          // down0
    pool(bufB, SK1, idx0, 0, 64);                   // -> (N1,64) skip1
    block(1, SK1,  64, prm[3], 128, bufB);          // down1
    pool(bufB, SK2, idx1, 1, 128);                  // -> (N2,128) skip2
    block(2, SK2, 128, prm[4], 256, bufB);          // down2
    pool(bufB, bufA, idx2, 2, 256);                 // -> (N3,256)
    block(3, bufA, 256, prm[5], 256, bufB);         // bott
    block(3, bufB, 256, prm[6], 256, bufA);         // pre0
    unpool_cat(bufA, idx2, SK2, bufB, 2, 256, 128); // -> (N2,384)
    block(2, bufB, 384, prm[7], 128, bufA);         // post0
    block(2, bufA, 128, prm[8], 128, bufB);         // pre1
    unpool_cat(bufB, idx1, SK1, bufA, 1, 128, 64);  // -> (N1,192)
    block(1, bufA, 192, prm[9], 64,  bufB);         // post1
    block(1, bufB, 64,  prm[10], 64, bufA);         // pre2
    unpool_cat(bufA, idx0, SK0, bufB, 0, 64, 32);   // -> (N0,96)
    block(0, bufB, 96,  prm[11], 32, bufA);         // post2
    cheb(0, bufA, 32, prm[12].W, 1, (float*)d_out); // outc (no BN/ReLU)
    int tot = 2 * NP[0];
    addbias_k<<<dim3((tot + 255) / 256), dim3(256), 0, stream>>>(
        (float*)d_out, prm[12].b, tot);
}